// DGCNN_15264313770135
// MI455X (gfx1250) — compile-verified
//
#include <hip/hip_runtime.h>

typedef __attribute__((ext_vector_type(2))) float v2f;
typedef __attribute__((ext_vector_type(8))) float v8f;

#define NNODES  131072
#define NPG     64
#define NGRAPH  2048
#define NEDGE   2097152
#define INDIM   128
#define HDIM    64
#define KTOP    30
#define MLPD    192
#define NC1     16

// ---------------- adjacency construction ----------------

__global__ void k_init_deg(float* __restrict__ deg) {
  int i = blockIdx.x * blockDim.x + threadIdx.x;
  if (i < NNODES) deg[i] = 1.0f;                    // self-loop
}

__global__ void k_deg(const int* __restrict__ ei, float* __restrict__ deg) {
  int e = blockIdx.x * blockDim.x + threadIdx.x;
  if (e < NEDGE) atomicAdd(&deg[ei[NEDGE + e]], 1.0f);
}

__global__ void k_rsqrt(float* __restrict__ deg) {
  int i = blockIdx.x * blockDim.x + threadIdx.x;
  if (i < NNODES) deg[i] = rsqrtf(deg[i]);          // deg -> dinv in place
}

__global__ void k_adj(const int* __restrict__ ei, const float* __restrict__ dinv,
                      float* __restrict__ A) {
  int e = blockIdx.x * blockDim.x + threadIdx.x;
  if (e < NEDGE) {
    int s = ei[e], d = ei[NEDGE + e];               // same graph by construction
    atomicAdd(&A[((s >> 6) << 12) + ((d & 63) << 6) + (s & 63)], dinv[s] * dinv[d]);
  }
}

__global__ void k_selfloop(const float* __restrict__ dinv, float* __restrict__ A) {
  int i = blockIdx.x * blockDim.x + threadIdx.x;
  if (i < NNODES) A[((i >> 6) << 12) + (i & 63) * 65] += dinv[i] * dinv[i];
}

// ---------------- fused per-graph forward ----------------
// D(64x64) = Asrc(64xK) @ B(Kx64); B supplied TRANSPOSED: Bt[n][k], ld = ldbt.
// Both fragments are contiguous float2 -> single ds_load_b64, no repack movs.
// A frag (16x4 f32): lane L -> row = L%16, k-pair base = (L/16)*2
// B frag (4x16 f32): lane L -> col = L%16, k-pair base = (L/16)*2
// C/D (16x16 f32):   VGPR v -> row = v + 8*(L/16), col = L%16
template <int K>
__device__ __forceinline__ void wmma_mm64(const float* __restrict__ Asrc, int lda,
                                          const float* __restrict__ Bt, int ldbt,
                                          float* __restrict__ Dst, bool storeT, int lddT,
                                          const float* __restrict__ bias, bool dorelu,
                                          int lane, int wave) {
  const int row16 = lane & 15;
  const int koff  = (lane >> 4) << 1;               // 0 or 2
  for (int it = 0; it < 2; ++it) {
    const int t  = wave * 2 + it;
    const int r0 = (t >> 2) << 4;
    const int c0 = (t & 3) << 4;
    v8f acc = {0.f, 0.f, 0.f, 0.f, 0.f, 0.f, 0.f, 0.f};
    const float* arow = Asrc + (r0 + row16) * lda + koff;   // even index: 8B aligned
    const float* brow = Bt + (c0 + row16) * ldbt + koff;    // ldbt even: 8B aligned
#pragma unroll
    for (int k = 0; k < K; k += 4) {
      v2f a = *(const v2f*)(arow + k);
      v2f b = *(const v2f*)(brow + k);
      acc = __builtin_amdgcn_wmma_f32_16x16x4_f32(false, a, false, b,
                                                  (short)0, acc, false, false);
    }
    const int cb   = c0 + row16;
    const int rb   = r0 + ((lane >> 4) << 3);
    const float bv = bias ? bias[cb] : 0.0f;
#pragma unroll
    for (int v = 0; v < 8; ++v) {
      float val = acc[v] + bv;
      if (dorelu) val = fmaxf(val, 0.0f);
      if (storeT) Dst[cb * lddT + rb + v] = val;    // transposed (B-side of next GEMM)
      else        Dst[(rb + v) * HDIM + cb] = val;  // row-major
    }
  }
}

#define LDW1 (INDIM + 2)   // 130: padded ld for transposed W1
#define LDT  (HDIM + 2)    // 66:  padded ld for transposed W2/W3 and T

__launch_bounds__(256)
__global__ void k_fused(const float* __restrict__ x, const float* __restrict__ A,
                        const float* __restrict__ W1, const float* __restrict__ b1,
                        const float* __restrict__ W2, const float* __restrict__ b2,
                        const float* __restrict__ W3, const float* __restrict__ b3,
                        const float* __restrict__ conv_w, const float* __restrict__ conv_b,
                        const float* __restrict__ fc1_w, const float* __restrict__ fc1_b,
                        const float* __restrict__ fc2_w, const float* __restrict__ fc2_b,
                        float* __restrict__ out) {
  __shared__ float smem[4096 + 8192 + HDIM * LDW1 + HDIM * LDT + 4096];  // ~113 KB
  __shared__ int   sIdx[KTOP];
  __shared__ float sConv[KTOP * NC1];
  __shared__ float sPool[NC1 * (KTOP / 2)];
  __shared__ float sFC1[128];

  float* sA  = smem;                                 // 4096: adjacency
  float* sB0 = smem + 4096;                          // 8192: X, later H1|H2
  float* sWt = smem + 12288;                         // 8320: transposed weights (reused)
  float* sTt = sWt + HDIM * LDW1;                    // 4224: transposed H@W temp
  float* sH3 = sTt + HDIM * LDT;                     // 4096
  float* sH1 = sB0;
  float* sH2 = sB0 + 4096;

  const int g    = blockIdx.x;
  const int tid  = threadIdx.x;
  const int lane = tid & 31;
  const int wave = tid >> 5;

  // stage X (contiguous), adjacency, and W1 transposed: Wt[n*ld + k] = W[k*64 + n]
  for (int i = tid; i < NPG * INDIM; i += 256) sB0[i] = x[(size_t)g * NPG * INDIM + i];
  for (int i = tid; i < NPG * NPG;   i += 256) sA[i]  = A[(size_t)g * NPG * NPG + i];
  for (int i = tid; i < INDIM * HDIM; i += 256) {
    int n = i / INDIM, k = i - n * INDIM;
    sWt[n * LDW1 + k] = W1[k * HDIM + n];
  }
  __syncthreads();

  wmma_mm64<INDIM>(sB0, INDIM, sWt, LDW1, sTt, true, LDT, nullptr, false, lane, wave);
  __syncthreads();
  wmma_mm64<HDIM>(sA, HDIM, sTt, LDT, sH1, false, 0, b1, true, lane, wave);
  __syncthreads();
  for (int i = tid; i < HDIM * HDIM; i += 256) {
    int n = i / HDIM, k = i - n * HDIM;
    sWt[n * LDT + k] = W2[k * HDIM + n];
  }
  __syncthreads();
  wmma_mm64<HDIM>(sH1, HDIM, sWt, LDT, sTt, true, LDT, nullptr, false, lane, wave);
  __syncthreads();
  wmma_mm64<HDIM>(sA, HDIM, sTt, LDT, sH2, false, 0, b2, true, lane, wave);
  __syncthreads();
  for (int i = tid; i < HDIM * HDIM; i += 256) {
    int n = i / HDIM, k = i - n * HDIM;
    sWt[n * LDT + k] = W3[k * HDIM + n];
  }
  __syncthreads();
  wmma_mm64<HDIM>(sH2, HDIM, sWt, LDT, sTt, true, LDT, nullptr, false, lane, wave);
  __syncthreads();
  wmma_mm64<HDIM>(sA, HDIM, sTt, LDT, sH3, false, 0, b3, true, lane, wave);
  __syncthreads();

  // ---- stable descending top-K on H3[:,63] (matches jnp.argsort(-key)) ----
  if (tid < NPG) {
    float ki = sH3[tid * HDIM + 63];
    int rank = 0;
    for (int j = 0; j < NPG; ++j) {
      float kj = sH3[j * HDIM + 63];
      rank += (kj > ki) || (kj == ki && j < tid);
    }
    if (rank < KTOP) sIdx[rank] = tid;
  }
  __syncthreads();

  // ---- conv1d(1->16, kernel=stride=192) over concat(H1,H2,H3) rows ----
  for (int p = tid; p < KTOP * NC1; p += 256) {
    int k = p / NC1, c = p % NC1;
    int node = sIdx[k];
    const float* w  = conv_w + c * MLPD;
    const float* f1 = sH1 + node * HDIM;
    const float* f2 = sH2 + node * HDIM;
    const float* f3 = sH3 + node * HDIM;
    float s = conv_b[c];
    for (int m = 0; m < HDIM; ++m) s += f1[m] * w[m];
    for (int m = 0; m < HDIM; ++m) s += f2[m] * w[HDIM + m];
    for (int m = 0; m < HDIM; ++m) s += f3[m] * w[2 * HDIM + m];
    sConv[k * NC1 + c] = fmaxf(s, 0.0f);
  }
  __syncthreads();

  // ---- channel-major maxpool over slot pairs ----
  for (int p = tid; p < NC1 * (KTOP / 2); p += 256) {
    int c = p / (KTOP / 2), j = p % (KTOP / 2);
    sPool[c * (KTOP / 2) + j] =
        fmaxf(sConv[(2 * j) * NC1 + c], sConv[(2 * j + 1) * NC1 + c]);
  }
  __syncthreads();

  // ---- fc1 (240 -> 128) + relu ----
  if (tid < 128) {
    float s = fc1_b[tid];
    for (int m = 0; m < NC1 * (KTOP / 2); ++m) s += sPool[m] * fc1_w[m * 128 + tid];
    sFC1[tid] = fmaxf(s, 0.0f);
  }
  __syncthreads();

  // ---- fc2 (128 -> 1) ----
  if (tid == 0) {
    float s = fc2_b[0];
    for (int m = 0; m < 128; ++m) s += sFC1[m] * fc2_w[m];
    out[g] = s;
  }
}

// ---------------- launch ----------------

extern "C" void kernel_launch(void* const* d_in, const int* in_sizes, int n_in,
                              void* d_out, int out_size, void* d_ws, size_t ws_size,
                              hipStream_t stream) {
  (void)in_sizes; (void)n_in; (void)out_size; (void)ws_size;
  const float* x      = (const float*)d_in[0];
  const int*   ei     = (const int*)d_in[1];        // [2, E] flat
  const float* W1     = (const float*)d_in[3];
  const float* b1     = (const float*)d_in[4];
  const float* W2     = (const float*)d_in[5];
  const float* b2     = (const float*)d_in[6];
  const float* W3     = (const float*)d_in[7];
  const float* b3     = (const float*)d_in[8];
  const float* conv_w = (const float*)d_in[9];
  const float* conv_b = (const float*)d_in[10];
  const float* fc1_w  = (const float*)d_in[11];
  const float* fc1_b  = (const float*)d_in[12];
  const float* fc2_w  = (const float*)d_in[13];
  const float* fc2_b  = (const float*)d_in[14];
  float* out = (float*)d_out;

  float* deg = (float*)d_ws;                        // N floats (deg -> dinv)
  float* A   = deg + NNODES;                        // G*64*64 floats (32 MB, L2-resident)

  hipMemsetAsync(A, 0, (size_t)NGRAPH * NPG * NPG * sizeof(float), stream);
  k_init_deg<<<NNODES / 256, 256, 0, stream>>>(deg);
  k_deg<<<NEDGE / 256, 256, 0, stream>>>(ei, deg);
  k_rsqrt<<<NNODES / 256, 256, 0, stream>>>(deg);
  k_adj<<<NEDGE / 256, 256, 0, stream>>>(ei, deg, A);
  k_selfloop<<<NNODES / 256, 256, 0, stream>>>(deg, A);
  k_fused<<<NGRAPH, 256, 0, stream>>>(x, A, W1, b1, W2, b2, W3, b3,
                                      conv_w, conv_b, fc1_w, fc1_b, fc2_w, fc2_b, out);
}